// DGCNN_76982993813720
// MI455X (gfx1250) — compile-verified
//
#include <hip/hip_runtime.h>

#define BATCH   4
#define NPT     4096
#define BN      (BATCH * NPT)        // 16384 points total
#define KNN     20
#define EF_ROWS (BN * KNN)           // 327680 (point,neighbor) rows

typedef __attribute__((ext_vector_type(2))) float v2f;
typedef __attribute__((ext_vector_type(8))) float v8f;

// =====================================================================
// WMMA f32 GEMM:  C(MxN) = A(MxK) * B^T   with B stored as (N,K) row-major.
// One wave computes a 32x64 tile (8 WMMA accumulators; B-fragments reused by
// both m-halves). K-loop is software-pipelined: next-iteration A/B fragments
// are prefetched into registers before the current WMMAs issue, so VMEM
// overlaps the matrix pipe instead of s_wait_loadcnt 0x0 before every WMMA.
// V_WMMA_F32_16X16X4_F32 layout: lanes0-15 -> K pair {k0,k0+1}, lanes16-31 ->
// {k0+2,k0+3}; C/D: VGPR r holds M=r (lanes0-15) / M=8+r (lanes16-31), N=lane&15.
// Epilogues: fused bias+ReLU, or fused pd_neg = 2*inner - sq[m] - sq[n].
// =====================================================================
template<bool BIAS, bool RELU, bool PDNEG>
__global__ __launch_bounds__(256) void k_gemm(
    const float* __restrict__ A, const float* __restrict__ BT,
    const float* __restrict__ bias, const float* __restrict__ sq,
    float* __restrict__ C, int M, int N, int K, int lda, int ldb, int ldc)
{
    const int lane    = threadIdx.x & 31;
    const int wave    = threadIdx.x >> 5;
    const int tiles_n = N >> 6;
    const int total   = (M >> 5) * tiles_n;
    const int tile    = blockIdx.x * 8 + wave;
    if (tile >= total) return;
    const int m0  = (tile / tiles_n) << 5;
    const int n0  = (tile % tiles_n) << 6;
    const int row = lane & 15;
    const int kk  = (lane >> 4) << 1;      // 0 or 2

    v8f acc[8] = {};
    const float* Ar0 = A + (size_t)(m0 + row) * lda + kk;
    const float* Ar1 = Ar0 + (size_t)16 * lda;
    const float* B0  = BT + (size_t)(n0 + row) * ldb + kk;

    v2f a0 = *(const v2f*)(Ar0);
    v2f a1 = *(const v2f*)(Ar1);
    v2f b[4];
#pragma unroll
    for (int t = 0; t < 4; ++t) b[t] = *(const v2f*)(B0 + (size_t)(t << 4) * ldb);

    int k0 = 0;
    for (; k0 + 4 < K; k0 += 4) {
        // prefetch next fragments (issues loads before current WMMAs wait)
        v2f a0n = *(const v2f*)(Ar0 + k0 + 4);
        v2f a1n = *(const v2f*)(Ar1 + k0 + 4);
        v2f bn[4];
#pragma unroll
        for (int t = 0; t < 4; ++t)
            bn[t] = *(const v2f*)(B0 + (size_t)(t << 4) * ldb + k0 + 4);
#pragma unroll
        for (int t = 0; t < 4; ++t) {
            acc[t]     = __builtin_amdgcn_wmma_f32_16x16x4_f32(
                false, a0, false, b[t], (short)0, acc[t], false, false);
            acc[4 + t] = __builtin_amdgcn_wmma_f32_16x16x4_f32(
                false, a1, false, b[t], (short)0, acc[4 + t], false, false);
        }
        a0 = a0n; a1 = a1n;
#pragma unroll
        for (int t = 0; t < 4; ++t) b[t] = bn[t];
    }
    // peeled tail iteration
#pragma unroll
    for (int t = 0; t < 4; ++t) {
        acc[t]     = __builtin_amdgcn_wmma_f32_16x16x4_f32(
            false, a0, false, b[t], (short)0, acc[t], false, false);
        acc[4 + t] = __builtin_amdgcn_wmma_f32_16x16x4_f32(
            false, a1, false, b[t], (short)0, acc[4 + t], false, false);
    }

    const int nn = lane & 15;
    const int mh = (lane >> 4) << 3;       // 0 or 8
#pragma unroll
    for (int h = 0; h < 2; ++h) {
#pragma unroll
        for (int t = 0; t < 4; ++t) {
            const int ncol = n0 + (t << 4) + nn;
            float bv = BIAS ? bias[ncol] : 0.0f;
#pragma unroll
            for (int r = 0; r < 8; ++r) {
                const int mrow = m0 + (h << 4) + mh + r;
                float v = acc[h * 4 + t][r];
                if (PDNEG) { v = 2.0f * v - sq[mrow] - sq[ncol]; }
                else       { v += bv; if (RELU) v = fmaxf(v, 0.0f); }
                C[(size_t)mrow * ldc + ncol] = v;
            }
        }
    }
}

// ---------------------------------------------------------------------
__global__ void k_extract_pts(const float* __restrict__ x, float* __restrict__ pts) {
    int i = blockIdx.x * 256 + threadIdx.x;
    if (i >= BN) return;
    pts[i * 3 + 0] = x[i * 6 + 0];
    pts[i * 3 + 1] = x[i * 6 + 1];
    pts[i * 3 + 2] = x[i * 6 + 2];
}

// kNN in 3-D: whole batch resident in LDS (48 KB of 320 KB), one thread per
// query row; register-resident top-20 via unrolled conditional selects.
__global__ __launch_bounds__(256) void k_knn3(const float* __restrict__ P, int* __restrict__ idx) {
    __shared__ float sp[NPT * 3];
    const int b = blockIdx.x >> 4;                    // 16 blocks per batch
    const int r = ((blockIdx.x & 15) << 8) + threadIdx.x;
    const float* Pb = P + (size_t)b * NPT * 3;
    for (int i = threadIdx.x; i < NPT * 3; i += 256) sp[i] = Pb[i];
    __syncthreads();
    const float px = sp[r * 3], py = sp[r * 3 + 1], pz = sp[r * 3 + 2];
    float bv[KNN]; int bi[KNN];
#pragma unroll
    for (int t = 0; t < KNN; ++t) { bv[t] = -3.402823466e38f; bi[t] = 0; }
    float minv = -3.402823466e38f; int minp = 0;
    for (int j = 0; j < NPT; ++j) {
        float dx = px - sp[j * 3], dy = py - sp[j * 3 + 1], dz = pz - sp[j * 3 + 2];
        float s = -(dx * dx + dy * dy + dz * dz);
        if (s > minv) {
#pragma unroll
            for (int t = 0; t < KNN; ++t) {
                bool sel = (t == minp);
                bv[t] = sel ? s : bv[t];
                bi[t] = sel ? j : bi[t];
            }
            minv = bv[0]; minp = 0;
#pragma unroll
            for (int t = 1; t < KNN; ++t) if (bv[t] < minv) { minv = bv[t]; minp = t; }
        }
    }
    int* o = idx + (size_t)(b * NPT + r) * KNN;
#pragma unroll
    for (int t = 0; t < KNN; ++t) o[t] = bi[t];
}

// top-20 over a precomputed 4096-score row (for 64-D kNN, one batch at a time)
__global__ __launch_bounds__(256) void k_topk_scores(const float* __restrict__ S, int* __restrict__ idx) {
    const int r = blockIdx.x * 256 + threadIdx.x;
    const float* row = S + (size_t)r * NPT;
    float bv[KNN]; int bi[KNN];
#pragma unroll
    for (int t = 0; t < KNN; ++t) { bv[t] = -3.402823466e38f; bi[t] = 0; }
    float minv = -3.402823466e38f; int minp = 0;
    for (int j = 0; j < NPT; ++j) {
        float s = row[j];
        if (s > minv) {
#pragma unroll
            for (int t = 0; t < KNN; ++t) {
                bool sel = (t == minp);
                bv[t] = sel ? s : bv[t];
                bi[t] = sel ? j : bi[t];
            }
            minv = bv[0]; minp = 0;
#pragma unroll
            for (int t = 1; t < KNN; ++t) if (bv[t] < minv) { minv = bv[t]; minp = t; }
        }
    }
    int* o = idx + (size_t)r * KNN;
#pragma unroll
    for (int t = 0; t < KNN; ++t) o[t] = bi[t];
}

// First-conv decomposition for 3-D inputs (K=3, done on VALU):
// U = pts*(Wtop - Wbot) + bias ; V = pts*Wbot       W is (6,64) row-major.
__global__ void k_uvw3(const float* __restrict__ P, const float* __restrict__ W,
                       const float* __restrict__ bias, float* __restrict__ U,
                       float* __restrict__ V) {
    int t = blockIdx.x * 256 + threadIdx.x;
    if (t >= BN * 64) return;
    int i = t >> 6, c = t & 63;
    float p0 = P[i * 3], p1 = P[i * 3 + 1], p2 = P[i * 3 + 2];
    float wt0 = W[c],        wt1 = W[64 + c],  wt2 = W[128 + c];
    float wb0 = W[192 + c],  wb1 = W[256 + c], wb2 = W[320 + c];
    V[t] = p0 * wb0 + p1 * wb1 + p2 * wb2;
    U[t] = p0 * (wt0 - wb0) + p1 * (wt1 - wb1) + p2 * (wt2 - wb2) + bias[c];
}

// Wdiff/Wbot (transposed to (N,K)) for the 128->64 first convs
__global__ void k_wdiff(const float* __restrict__ W, float* __restrict__ WdT,
                        float* __restrict__ WbT) {
    int t = blockIdx.x * 256 + threadIdx.x;
    if (t >= 64 * 64) return;
    int k = t >> 6, c = t & 63;
    float top = W[k * 64 + c], bot = W[(k + 64) * 64 + c];
    WdT[c * 64 + k] = top - bot;
    WbT[c * 64 + k] = bot;
}

// h[(i,k),c] = relu(U[i,c] + V[idx[i,k],c]) — 64 channels, float4 vectorized
__global__ void k_gather_relu(const float* __restrict__ U, const float* __restrict__ V,
                              const int* __restrict__ idx, float* __restrict__ H) {
    int t = blockIdx.x * 256 + threadIdx.x;
    if (t >= EF_ROWS * 16) return;
    int c4 = t & 15;
    int ik = t >> 4;
    int i  = ik / KNN;
    int j  = ((i >> 12) << 12) + idx[ik];       // batch-local -> global row
    float4 u = ((const float4*)(U + (size_t)i * 64))[c4];
    float4 v = ((const float4*)(V + (size_t)j * 64))[c4];
    float4 o;
    o.x = fmaxf(u.x + v.x, 0.0f); o.y = fmaxf(u.y + v.y, 0.0f);
    o.z = fmaxf(u.z + v.z, 0.0f); o.w = fmaxf(u.w + v.w, 0.0f);
    ((float4*)(H + (size_t)ik * 64))[c4] = o;
}

__global__ void k_maxk(const float* __restrict__ H, float* __restrict__ out, int C, int total) {
    int t = blockIdx.x * 256 + threadIdx.x;
    if (t >= total) return;
    int i = t / C, c = t - i * C;
    const float* h = H + (size_t)i * KNN * C + c;
    float m = h[0];
#pragma unroll
    for (int k = 1; k < KNN; ++k) m = fmaxf(m, h[(size_t)k * C]);
    out[(size_t)i * C + c] = m;
}

// max over the 4096 points of one batch; grid = (BATCH, C/256)
__global__ void k_maxn(const float* __restrict__ in, float* __restrict__ out, int C) {
    int b = blockIdx.x;
    int c = blockIdx.y * 256 + threadIdx.x;
    if (c >= C) return;
    const float* p = in + ((size_t)b * NPT) * C + c;
    float m = p[0];
    for (int n = 1; n < NPT; ++n) m = fmaxf(m, p[(size_t)n * C]);
    out[(size_t)b * C + c] = m;
}

__global__ void k_sqnorm(const float* __restrict__ Xf, float* __restrict__ sq) {
    int i = blockIdx.x * 256 + threadIdx.x;
    if (i >= BN) return;
    const float* r = Xf + (size_t)i * 64;
    float s = 0.0f;
#pragma unroll
    for (int c = 0; c < 64; ++c) s += r[c] * r[c];
    sq[i] = s;
}

__global__ void k_transpose(const float* __restrict__ Win, float* __restrict__ WT, int K, int N) {
    int t = blockIdx.x * 256 + threadIdx.x;
    if (t >= K * N) return;
    int k = t / N, n = t - k * N;
    WT[(size_t)n * K + k] = Win[t];
}

__global__ void k_fc(const float* __restrict__ A, const float* __restrict__ Wt,
                     const float* __restrict__ bias, float* __restrict__ out,
                     int M, int K, int N, int relu) {
    int t = blockIdx.x * 256 + threadIdx.x;
    if (t >= M * N) return;
    int m = t / N, n = t - m * N;
    float s = bias[n];
    for (int k = 0; k < K; ++k) s += A[(size_t)m * K + k] * Wt[(size_t)k * N + n];
    out[t] = relu ? fmaxf(s, 0.0f) : s;
}

__global__ void k_transform(const float* __restrict__ pts, const float* __restrict__ xf,
                            float* __restrict__ p) {
    int i = blockIdx.x * 256 + threadIdx.x;
    if (i >= BN) return;
    const float* T = xf + (size_t)(i >> 12) * 9;
    float a = pts[i * 3], b = pts[i * 3 + 1], c = pts[i * 3 + 2];
#pragma unroll
    for (int d = 0; d < 3; ++d)
        p[i * 3 + d] = a * T[d] + b * T[3 + d] + c * T[6 + d];
}

__global__ void k_cat1(const float* __restrict__ x1, const float* __restrict__ x2,
                       const float* __restrict__ x3, float* __restrict__ out) {
    int t = blockIdx.x * 256 + threadIdx.x;
    if (t >= BN * 192) return;
    int i = t / 192, c = t - i * 192;
    float v = (c < 64) ? x1[(size_t)i * 64 + c]
            : (c < 128) ? x2[(size_t)i * 64 + c - 64]
                        : x3[(size_t)i * 64 + c - 128];
    out[t] = v;
}

__global__ void k_cat2(const float* __restrict__ g, const float* __restrict__ x1,
                       const float* __restrict__ x2, const float* __restrict__ x3,
                       float* __restrict__ out) {
    int t = blockIdx.x * 256 + threadIdx.x;
    if (t >= BN * 1216) return;
    int i = t / 1216, c = t - i * 1216;
    float v;
    if (c < 1024) v = g[(size_t)(i >> 12) * 1024 + c];
    else {
        int c2 = c - 1024;
        v = (c2 < 64) ? x1[(size_t)i * 64 + c2]
          : (c2 < 128) ? x2[(size_t)i * 64 + c2 - 64]
                       : x3[(size_t)i * 64 + c2 - 128];
    }
    out[t] = v;
}

__global__ void k_out2(const float* __restrict__ A, const float* __restrict__ Wt,
                       const float* __restrict__ bias, float* __restrict__ out) {
    int t = blockIdx.x * 256 + threadIdx.x;
    if (t >= BN * 2) return;
    int i = t >> 1, oc = t & 1;
    const float* r = A + (size_t)i * 256;
    float s = bias[oc];
    for (int k = 0; k < 256; ++k) s += r[k] * Wt[k * 2 + oc];
    out[t] = s;
}

// =====================================================================
extern "C" void kernel_launch(void* const* d_in, const int* in_sizes, int n_in,
                              void* d_out, int out_size, void* d_ws, size_t ws_size,
                              hipStream_t stream) {
    const float* X    = (const float*)d_in[0];
    const float* TW1  = (const float*)d_in[1];  const float* TB1 = (const float*)d_in[2];
    const float* TW2  = (const float*)d_in[3];  const float* TB2 = (const float*)d_in[4];
    const float* TW3  = (const float*)d_in[5];  const float* TB3 = (const float*)d_in[6];
    const float* TFW1 = (const float*)d_in[7];  const float* TFB1 = (const float*)d_in[8];
    const float* TFW2 = (const float*)d_in[9];  const float* TFB2 = (const float*)d_in[10];
    const float* TXW  = (const float*)d_in[11]; const float* TXB = (const float*)d_in[12];
    const float* E1W1 = (const float*)d_in[13]; const float* E1B1 = (const float*)d_in[14];
    const float* E1W2 = (const float*)d_in[15]; const float* E1B2 = (const float*)d_in[16];
    const float* E2W1 = (const float*)d_in[17]; const float* E2B1 = (const float*)d_in[18];
    const float* E2W2 = (const float*)d_in[19]; const float* E2B2 = (const float*)d_in[20];
    const float* E3W1 = (const float*)d_in[21]; const float* E3B1 = (const float*)d_in[22];
    const float* C1W  = (const float*)d_in[23]; const float* C1B = (const float*)d_in[24];
    const float* C2W  = (const float*)d_in[25]; const float* C2B = (const float*)d_in[26];
    const float* C3W  = (const float*)d_in[27]; const float* C3B = (const float*)d_in[28];
    const float* C4W  = (const float*)d_in[29]; const float* C4B = (const float*)d_in[30];
    float* OUT = (float*)d_out;

    // ---- workspace layout (regions reused across lifetimes; ~292 MB peak) ----
    char* Wp = (char*)d_ws;
    size_t off = 0;
    auto take = [&](size_t bytes) {
        void* p = Wp + off;
        off += (bytes + 255) & ~(size_t)255;
        return (float*)p;
    };
    float* RA  = take(167772160ULL);  // 168MB: t2 / e-conv2 out / inner / net1 / net2
    float* RB  = take(83886080ULL);   //  84MB: gathered h / cat2 / net3
    float* RD  = take(12582912ULL);   //  12MB: tmax / cat1
    float* PTS = take((size_t)BN * 3 * 4);
    float* PP  = take((size_t)BN * 3 * 4);
    int*   IDX = (int*)take((size_t)BN * KNN * 4);
    float* UB  = take((size_t)BN * 64 * 4);
    float* VB  = take((size_t)BN * 64 * 4);
    float* X1  = take((size_t)BN * 64 * 4);
    float* X2  = take((size_t)BN * 64 * 4);
    float* X3  = take((size_t)BN * 64 * 4);
    float* SQ  = take((size_t)BN * 4);
    float* G   = take((size_t)BATCH * 1024 * 4);
    float* FC1 = take((size_t)BATCH * 512 * 4);
    float* FC2 = take((size_t)BATCH * 256 * 4);
    float* XF  = take((size_t)BATCH * 9 * 4);
    float* TW2T  = take(64 * 128 * 4);
    float* TW3T  = take(128 * 1024 * 4);
    float* E1W2T = take(64 * 64 * 4);
    float* E2W2T = take(64 * 64 * 4);
    float* C1WT  = take(192 * 1024 * 4);
    float* C2WT  = take((size_t)1216 * 512 * 4);
    float* C3WT  = take(512 * 256 * 4);
    float* WDT   = take(64 * 64 * 4);
    float* WBT   = take(64 * 64 * 4);

    auto gblk = [](int M, int N) { return ((M >> 5) * (N >> 6) + 7) / 8; };
    auto tblk = [](int n) { return (n + 255) / 256; };

    // ---- weight transposes to (N,K) for contiguous WMMA B-fragments ----
    k_transpose<<<tblk(64 * 128), 256, 0, stream>>>(TW2, TW2T, 64, 128);
    k_transpose<<<tblk(128 * 1024), 256, 0, stream>>>(TW3, TW3T, 128, 1024);
    k_transpose<<<tblk(64 * 64), 256, 0, stream>>>(E1W2, E1W2T, 64, 64);
    k_transpose<<<tblk(64 * 64), 256, 0, stream>>>(E2W2, E2W2T, 64, 64);
    k_transpose<<<tblk(192 * 1024), 256, 0, stream>>>(C1W, C1WT, 192, 1024);
    k_transpose<<<tblk(1216 * 512), 256, 0, stream>>>(C2W, C2WT, 1216, 512);
    k_transpose<<<tblk(512 * 256), 256, 0, stream>>>(C3W, C3WT, 512, 256);

    // ---- stage A: spatial transformer net ----
    k_extract_pts<<<64, 256, 0, stream>>>(X, PTS);
    k_knn3<<<64, 256, 0, stream>>>(PTS, IDX);
    k_uvw3<<<tblk(BN * 64), 256, 0, stream>>>(PTS, TW1, TB1, UB, VB);
    k_gather_relu<<<tblk(EF_ROWS * 16), 256, 0, stream>>>(UB, VB, IDX, RB);
    k_gemm<true, true, false><<<gblk(EF_ROWS, 128), 256, 0, stream>>>(
        RB, TW2T, TB2, nullptr, RA, EF_ROWS, 128, 64, 64, 64, 128);          // t2
    k_maxk<<<tblk(BN * 128), 256, 0, stream>>>(RA, RD, 128, BN * 128);       // tmax
    k_gemm<true, true, false><<<gblk(BN, 1024), 256, 0, stream>>>(
        RD, TW3T, TB3, nullptr, RA, BN, 1024, 128, 128, 128, 1024);          // t3
    k_maxn<<<dim3(BATCH, 4), 256, 0, stream>>>(RA, G, 1024);                 // tglob
    k_fc<<<tblk(BATCH * 512), 256, 0, stream>>>(G, TFW1, TFB1, FC1, BATCH, 1024, 512, 1);
    k_fc<<<tblk(BATCH * 256), 256, 0, stream>>>(FC1, TFW2, TFB2, FC2, BATCH, 512, 256, 1);
    k_fc<<<1, 64, 0, stream>>>(FC2, TXW, TXB, XF, BATCH, 256, 9, 0);
    k_transform<<<64, 256, 0, stream>>>(PTS, XF, PP);

    // ---- stage B1: EdgeConv 1 (3-D input) ----
    k_knn3<<<64, 256, 0, stream>>>(PP, IDX);
    k_uvw3<<<tblk(BN * 64), 256, 0, stream>>>(PP, E1W1, E1B1, UB, VB);
    k_gather_relu<<<tblk(EF_ROWS * 16), 256, 0, stream>>>(UB, VB, IDX, RB);
    k_gemm<true, true, false><<<gblk(EF_ROWS, 64), 256, 0, stream>>>(
        RB, E1W2T, E1B2, nullptr, RA, EF_ROWS, 64, 64, 64, 64, 64);
    k_maxk<<<tblk(BN * 64), 256, 0, stream>>>(RA, X1, 64, BN * 64);          // x1

    // ---- stage B2: EdgeConv 2 (64-D kNN via WMMA pd_neg) ----
    k_sqnorm<<<64, 256, 0, stream>>>(X1, SQ);
    for (int b = 0; b < BATCH; ++b) {
        const float* Xb = X1 + (size_t)b * NPT * 64;
        k_gemm<false, false, true><<<gblk(NPT, NPT), 256, 0, stream>>>(
            Xb, Xb, nullptr, SQ + (size_t)b * NPT, RA, NPT, NPT, 64, 64, 64, NPT);
        k_topk_scores<<<16, 256, 0, stream>>>(RA, IDX + (size_t)b * NPT * KNN);
    }
    k_wdiff<<<16, 256, 0, stream>>>(E2W1, WDT, WBT);
    k_gemm<true, false, false><<<gblk(BN, 64), 256, 0, stream>>>(
        X1, WDT, E2B1, nullptr, UB, BN, 64, 64, 64, 64, 64);                 // U2
    k_gemm<false, false, false><<<gblk(BN, 64), 256, 0, stream>>>(
        X1, WBT, nullptr, nullptr, VB, BN, 64, 64, 64, 64, 64);              // V2
    k_gather_relu<<<tblk(EF_ROWS * 16), 256, 0, stream>>>(UB, VB, IDX, RB);
    k_gemm<true, true, false><<<gblk(EF_ROWS, 64), 256, 0, stream>>>(
        RB, E2W2T, E2B2, nullptr, RA, EF_ROWS, 64, 64, 64, 64, 64);
    k_maxk<<<tblk(BN * 64), 256, 0, stream>>>(RA, X2, 64, BN * 64);          // x2

    // ---- stage B3: EdgeConv 3 ----
    k_sqnorm<<<64, 256, 0, stream>>>(X2, SQ);
    for (int b = 0; b < BATCH; ++b) {
        const float* Xb = X2 + (size_t)b * NPT * 64;
        k_gemm<false, false, true><<<gblk(NPT, NPT), 256, 0, stream>>>(
            Xb, Xb, nullptr, SQ + (size_t)b * NPT, RA, NPT, NPT, 64, 64, 64, NPT);
        k_topk_scores<<<16, 256, 0, stream>>>(RA, IDX + (size_t)b * NPT * KNN);
    }
    k_wdiff<<<16, 256, 0, stream>>>(E3W1, WDT, WBT);
    k_gemm<true, false, false><<<gblk(BN, 64), 256, 0, stream>>>(
        X2, WDT, E3B1, nullptr, UB, BN, 64, 64, 64, 64, 64);                 // U3
    k_gemm<false, false, false><<<gblk(BN, 64), 256, 0, stream>>>(
        X2, WBT, nullptr, nullptr, VB, BN, 64, 64, 64, 64, 64);              // V3
    k_gather_relu<<<tblk(EF_ROWS * 16), 256, 0, stream>>>(UB, VB, IDX, RB);
    k_maxk<<<tblk(BN * 64), 256, 0, stream>>>(RB, X3, 64, BN * 64);          // x3

    // ---- stage C: head ----
    k_cat1<<<tblk(BN * 192), 256, 0, stream>>>(X1, X2, X3, RD);
    k_gemm<true, true, false><<<gblk(BN, 1024), 256, 0, stream>>>(
        RD, C1WT, C1B, nullptr, RA, BN, 1024, 192, 192, 192, 1024);          // net1
    k_maxn<<<dim3(BATCH, 4), 256, 0, stream>>>(RA, G, 1024);                 // g
    k_cat2<<<tblk(BN * 1216), 256, 0, stream>>>(G, X1, X2, X3, RB);
    k_gemm<true, true, false><<<gblk(BN, 512), 256, 0, stream>>>(
        RB, C2WT, C2B, nullptr, RA, BN, 512, 1216, 1216, 1216, 512);         // net2
    k_gemm<true, true, false><<<gblk(BN, 256), 256, 0, stream>>>(
        RA, C3WT, C3B, nullptr, RB, BN, 256, 512, 512, 512, 256);            // net3
    k_out2<<<tblk(BN * 2), 256, 0, stream>>>(RB, C4W, C4B, OUT);
}